// LengthRegulator_27711128993922
// MI455X (gfx1250) — compile-verified
//
#include <hip/hip_runtime.h>
#include <hip/hip_bf16.h>

// LengthRegulator: out[b, j, :] = x[b, searchsorted(cumsum(dur[b]), j, 'right'), :]
// for j < out_lens[b], else 0.  Plus out_lens appended as int32 at the tail of d_out.

constexpr int kB = 32;
constexpr int kT = 1024;
constexpr int kD = 384;            // floats per row
constexpr int kV4 = kD / 4;        // 96 float4 per row
constexpr int ROWS_PER_BLOCK = 64;
constexpr int THREADS = 256;       // 8 wave32s

typedef float __attribute__((ext_vector_type(4))) v4f;
typedef int   __attribute__((ext_vector_type(4))) v4i;

// ---- CDNA5 async global<->LDS DMA path (gfx1250), guarded by __has_builtin ----
#if defined(__gfx1250__) && \
    __has_builtin(__builtin_amdgcn_global_load_async_to_lds_b128) && \
    __has_builtin(__builtin_amdgcn_global_store_async_from_lds_b128)
#define HAVE_ASYNC 1
#else
#define HAVE_ASYNC 0
#endif

// CPol immediates (gfx12-style: TH[2:0], SCOPE[4:3]).  Load x with RT so the
// 48 MB source stays L2-resident (reused ~3.5x); store output NT so the
// ~180 MB stream doesn't evict it.
#define CPOL_RT 0
#define CPOL_NT 1

#if HAVE_ASYNC
typedef __attribute__((address_space(1))) v4i gv4i;   // global int4 ("__device__ int4*")
typedef __attribute__((address_space(3))) v4i lv4i;   // LDS int4    ("__shared__ int4*")
__device__ __forceinline__ gv4i* to_g(const void* p) { return (gv4i*)p; }
__device__ __forceinline__ lv4i* to_l(const void* p) { return (lv4i*)p; }
#if __has_builtin(__builtin_amdgcn_s_wait_asynccnt)
#define WAIT_ASYNC0() __builtin_amdgcn_s_wait_asynccnt(0)
#else
#define WAIT_ASYNC0() asm volatile("s_wait_asynccnt 0" ::: "memory")
#endif
#endif

__global__ __launch_bounds__(THREADS)
void lenreg_kernel(const float* __restrict__ x, const int* __restrict__ dur,
                   float* __restrict__ out, int* __restrict__ out_lens,
                   int max_len) {
  __shared__ int s_cum[kT];
  __shared__ int s_part[THREADS];
#if HAVE_ASYNC
  __shared__ v4f s_stage[THREADS / 32][kV4];   // 12 KB staging, one row per wave
#endif

  const int t = threadIdx.x;
  const int b = blockIdx.y;

  // ---- block-local inclusive scan of durations[b][0..1023] (4 elems/thread) ----
  const int4 d4 = ((const int4*)(dur + (size_t)b * kT))[t];
  const int s0 = d4.x;
  const int s1 = s0 + d4.y;
  const int s2 = s1 + d4.z;
  const int s3 = s2 + d4.w;
  s_part[t] = s3;
  __syncthreads();
#pragma unroll
  for (int off = 1; off < THREADS; off <<= 1) {
    const int v = (t >= off) ? s_part[t - off] : 0;
    __syncthreads();
    s_part[t] += v;
    __syncthreads();
  }
  const int excl = (t == 0) ? 0 : s_part[t - 1];
  s_cum[4 * t + 0] = excl + s0;
  s_cum[4 * t + 1] = excl + s1;
  s_cum[4 * t + 2] = excl + s2;
  s_cum[4 * t + 3] = excl + s3;
  __syncthreads();
  const int total = s_part[THREADS - 1];   // out_lens[b]

  if (blockIdx.x == 0 && t == 0) out_lens[b] = total;

  const int warp = t >> 5;
  const int lane = t & 31;
  const int row0 = blockIdx.x * ROWS_PER_BLOCK;
  const int rowEnd = (row0 + ROWS_PER_BLOCK < max_len) ? row0 + ROWS_PER_BLOCK : max_len;

  for (int j = row0 + warp; j < rowEnd; j += THREADS / 32) {
    // searchsorted(cum, j, 'right'): smallest idx with cum[idx] > j
    int lo = 0, hi = kT;
    while (lo < hi) {
      const int mid = (lo + hi) >> 1;
      if (s_cum[mid] > j) hi = mid; else lo = mid + 1;
    }
    const int idx = (lo < kT) ? lo : kT - 1;

    v4f* dst = (v4f*)(out + ((size_t)b * max_len + j) * kD);
    if (j < total) {
      const v4f* src = (const v4f*)(x + ((size_t)b * kT + idx) * kD);
#if HAVE_ASYNC
      // DMA row memory -> LDS -> memory, no VGPR round-trip (ASYNCcnt tracked).
      WAIT_ASYNC0();                       // prior stores from this wave's stage done
#pragma unroll
      for (int k = 0; k < kV4 / 32; ++k)
        __builtin_amdgcn_global_load_async_to_lds_b128(
            to_g(src + lane + 32 * k), to_l(&s_stage[warp][lane + 32 * k]), 0, CPOL_RT);
      WAIT_ASYNC0();                       // loads landed in LDS
#pragma unroll
      for (int k = 0; k < kV4 / 32; ++k)
        __builtin_amdgcn_global_store_async_from_lds_b128(
            to_g(dst + lane + 32 * k), to_l(&s_stage[warp][lane + 32 * k]), 0, CPOL_NT);
#else
#pragma unroll
      for (int k = 0; k < kV4 / 32; ++k) {
        const v4f v = src[lane + 32 * k];                 // x is L2-resident (48 MB)
        __builtin_nontemporal_store(v, dst + lane + 32 * k);  // stream output, keep x in L2
      }
#endif
    } else {
      const v4f z = (v4f)0.f;
#pragma unroll
      for (int k = 0; k < kV4 / 32; ++k)
        __builtin_nontemporal_store(z, dst + lane + 32 * k);
    }
  }
#if HAVE_ASYNC
  WAIT_ASYNC0();   // drain before wave exit (S_ENDPGM would wait anyway)
#endif
}

extern "C" void kernel_launch(void* const* d_in, const int* in_sizes, int n_in,
                              void* d_out, int out_size, void* d_ws, size_t ws_size,
                              hipStream_t stream) {
  (void)in_sizes; (void)n_in; (void)d_ws; (void)ws_size;
  const float* x  = (const float*)d_in[0];
  const int* dur  = (const int*)d_in[1];
  float* out      = (float*)d_out;

  // out_size = B*max_len*D + B  (tuple concatenated flat)
  const int max_len = (out_size - kB) / (kB * kD);
  if (max_len <= 0) return;
  int* out_lens = (int*)d_out + (size_t)kB * max_len * kD;

  dim3 grid((max_len + ROWS_PER_BLOCK - 1) / ROWS_PER_BLOCK, kB);
  lenreg_kernel<<<grid, THREADS, 0, stream>>>(x, dur, out, out_lens, max_len);
}